// TemporalAttnDecoderRNN_31714038514462
// MI455X (gfx1250) — compile-verified
//
#include <hip/hip_runtime.h>
#include <math.h>

constexpr int BB = 64;
constexpr int TT = 50;
constexpr int SS = 400;
constexpr int HH = 512;
constexpr int EE = 256;
constexpr int VV = 50000;
constexpr int H3 = 3 * HH;

typedef float v2f __attribute__((ext_vector_type(2)));
typedef float v8f __attribute__((ext_vector_type(8)));

__device__ __forceinline__ float sigmoidf_(float x) { return 1.0f / (1.0f + expf(-x)); }

// ---------------- K1: embedding gather + GRU step -> h (B,H) ----------------
__global__ void k_gru(const int* __restrict__ tok, const float* __restrict__ emb,
                      const float* __restrict__ last_h,
                      const float* __restrict__ W_ih, const float* __restrict__ W_hh,
                      const float* __restrict__ b_ih, const float* __restrict__ b_hh,
                      float* __restrict__ h_out) {
  int idx = blockIdx.x * blockDim.x + threadIdx.x;
  if (idx >= BB * HH) return;
  int b = idx / HH, j = idx - b * HH;
  const float* x  = emb + (size_t)tok[b] * EE;
  const float* hp = last_h + (size_t)b * HH;
  float ir = b_ih[j], iz = b_ih[HH + j], inn = b_ih[2 * HH + j];
  const float* w0 = W_ih + (size_t)j * EE;
  const float* w1 = W_ih + (size_t)(HH + j) * EE;
  const float* w2 = W_ih + (size_t)(2 * HH + j) * EE;
  for (int e = 0; e < EE; ++e) {
    float xe = x[e];
    ir = fmaf(xe, w0[e], ir); iz = fmaf(xe, w1[e], iz); inn = fmaf(xe, w2[e], inn);
  }
  float hr = b_hh[j], hz = b_hh[HH + j], hn = b_hh[2 * HH + j];
  const float* u0 = W_hh + (size_t)j * HH;
  const float* u1 = W_hh + (size_t)(HH + j) * HH;
  const float* u2 = W_hh + (size_t)(2 * HH + j) * HH;
  for (int k = 0; k < HH; ++k) {
    float hh = hp[k];
    hr = fmaf(hh, u0[k], hr); hz = fmaf(hh, u1[k], hz); hn = fmaf(hh, u2[k], hn);
  }
  float r = sigmoidf_(ir + hr);
  float z = sigmoidf_(iz + hz);
  float n = tanhf(inn + r * hn);
  h_out[idx] = (1.0f - z) * n + z * hp[j];
}

// ---------------- K2a: att_dist + temporal (warp per (b,s)) ----------------
__global__ void k_attdist(const float* __restrict__ h, const float* __restrict__ w_h,
                          const float* __restrict__ enc, const float* __restrict__ prev_att,
                          float* __restrict__ att_dist, float* __restrict__ temporal) {
  int gw = (blockIdx.x * blockDim.x + threadIdx.x) >> 5;
  int lane = threadIdx.x & 31;
  if (gw >= BB * SS) return;
  int b = gw / SS, s = gw - b * SS;
  const float* hr = h + (size_t)b * HH;
  const float* er = enc + ((size_t)b * SS + s) * HH;
  float acc = 0.f;
  for (int k = lane; k < HH; k += 32) acc = fmaf(hr[k] * w_h[k], er[k], acc);
  for (int off = 16; off > 0; off >>= 1) acc += __shfl_xor(acc, off, 32);
  float den = 0.f;
  for (int t = lane; t < TT; t += 32) den += expf(prev_att[((size_t)b * TT + t) * SS + s]);
  for (int off = 16; off > 0; off >>= 1) den += __shfl_xor(den, off, 32);
  if (lane == 0) {
    att_dist[(size_t)b * SS + s] = acc;
    temporal[(size_t)b * SS + s] = expf(acc) / den;
  }
}

// ---------------- K2b: normalize temporal -> att ----------------
__global__ void k_attnorm(const float* __restrict__ temporal, float* __restrict__ att) {
  int b = blockIdx.x, tid = threadIdx.x;
  __shared__ float red[256];
  float p = 0.f;
  for (int s = tid; s < SS; s += 256) p += temporal[(size_t)b * SS + s];
  red[tid] = p; __syncthreads();
  for (int off = 128; off > 0; off >>= 1) { if (tid < off) red[tid] += red[tid + off]; __syncthreads(); }
  float inv = 1.0f / red[0];
  for (int s = tid; s < SS; s += 256) att[(size_t)b * SS + s] = temporal[(size_t)b * SS + s] * inv;
}

// ---------------- K3: dec attention, contexts, combined, p_gen ----------------
__global__ void k_ctx(const float* __restrict__ h, const float* __restrict__ att,
                      const float* __restrict__ enc, const float* __restrict__ pdh,
                      const float* __restrict__ w_d,
                      const float* __restrict__ W_gen, const float* __restrict__ b_gen,
                      float* __restrict__ combined, float* __restrict__ p_gen) {
  int b = blockIdx.x, tid = threadIdx.x;
  __shared__ float sc[TT];
  __shared__ float red[256];
  const float* hr = h + (size_t)b * HH;
  for (int t = tid; t < TT; t += 256) {
    const float* dr = pdh + ((size_t)b * TT + t) * HH;
    float a = 0.f;
    for (int k = 0; k < HH; ++k) a = fmaf(hr[k] * w_d[k], dr[k], a);
    sc[t] = a;
  }
  __syncthreads();
  if (tid == 0) {
    float m = -1e30f;
    for (int t = 0; t < TT; ++t) m = fmaxf(m, sc[t]);
    float sum = 0.f;
    for (int t = 0; t < TT; ++t) { sc[t] = expf(sc[t] - m); sum += sc[t]; }
    float inv = 1.0f / sum;
    for (int t = 0; t < TT; ++t) sc[t] *= inv;
  }
  __syncthreads();
  float* comb = combined + (size_t)b * H3;
  for (int j = tid; j < HH; j += 256) {
    comb[j] = hr[j];
    float ec = 0.f;
    for (int s = 0; s < SS; ++s) ec = fmaf(att[(size_t)b * SS + s], enc[((size_t)b * SS + s) * HH + j], ec);
    comb[HH + j] = ec;
    float dc = 0.f;
    for (int t = 0; t < TT; ++t) dc = fmaf(sc[t], pdh[((size_t)b * TT + t) * HH + j], dc);
    comb[2 * HH + j] = dc;
  }
  __syncthreads();
  float pg = 0.f;
  for (int i = tid; i < H3; i += 256) pg = fmaf(comb[i], W_gen[i], pg);
  red[tid] = pg; __syncthreads();
  for (int off = 128; off > 0; off >>= 1) { if (tid < off) red[tid] += red[tid + off]; __syncthreads(); }
  if (tid == 0) p_gen[b] = sigmoidf_(red[0] + b_gen[0]);
}

// ---------------- K4: ohe = combined @ W_oh.T + b_oh ----------------
__global__ void k_oh(const float* __restrict__ combined, const float* __restrict__ W_oh,
                     const float* __restrict__ b_oh, float* __restrict__ ohe) {
  int idx = blockIdx.x * blockDim.x + threadIdx.x;
  if (idx >= BB * EE) return;
  int b = idx / EE, e = idx - b * EE;
  const float* comb = combined + (size_t)b * H3;
  const float* w = W_oh + (size_t)e * H3;
  float a = b_oh[e];
  for (int i = 0; i < H3; ++i) a = fmaf(comb[i], w[i], a);
  ohe[idx] = a;
}

// ---------------- K5: logits via V_WMMA_F32_16X16X4_F32 ----------------
// One wave per 16-wide N-tile of V; each wave computes ALL 4 M-tiles (B=64),
// reusing each loaded W_ov (B) fragment across 4 WMMAs to quarter the W_ov
// load traffic (the dominant HBM stream). K=256 in 64 steps of 4.
__global__ __launch_bounds__(32) void k_logits(const float* __restrict__ ohe,
                                               const float* __restrict__ W_ov,
                                               const float* __restrict__ b_ov,
                                               float* __restrict__ logits) {
  int v0 = blockIdx.x * 16;           // N-tile origin in V
  int lane = threadIdx.x;
  int half = lane >> 4;               // K-pair selector per ISA A/B layout
  int mn = lane & 15;                 // lane's M (for A) and N (for B/D)
  const float* brow = W_ov + (size_t)(v0 + mn) * EE;   // B[k][n] = W_ov[v0+n][k]
  const float* a0 = ohe + (size_t)(0  + mn) * EE;
  const float* a1 = ohe + (size_t)(16 + mn) * EE;
  const float* a2 = ohe + (size_t)(32 + mn) * EE;
  const float* a3 = ohe + (size_t)(48 + mn) * EE;
  v8f c0 = {0.f,0.f,0.f,0.f,0.f,0.f,0.f,0.f};
  v8f c1 = c0, c2 = c0, c3 = c0;
  for (int k = 0; k < EE; k += 4) {
    int kk = k + half * 2;
    v2f bf, a;
    bf.x = brow[kk]; bf.y = brow[kk + 1];
    a.x = a0[kk]; a.y = a0[kk + 1];
    c0 = __builtin_amdgcn_wmma_f32_16x16x4_f32(false, a, false, bf, (short)0, c0, false, false);
    a.x = a1[kk]; a.y = a1[kk + 1];
    c1 = __builtin_amdgcn_wmma_f32_16x16x4_f32(false, a, false, bf, (short)0, c1, false, false);
    a.x = a2[kk]; a.y = a2[kk + 1];
    c2 = __builtin_amdgcn_wmma_f32_16x16x4_f32(false, a, false, bf, (short)0, c2, false, false);
    a.x = a3[kk]; a.y = a3[kk + 1];
    c3 = __builtin_amdgcn_wmma_f32_16x16x4_f32(false, a, false, bf, (short)0, c3, false, false);
  }
  float bias = b_ov[v0 + mn];
  // D layout: lanes 0-15 hold M = r (N = lane), lanes 16-31 hold M = 8 + r.
#pragma unroll
  for (int r = 0; r < 8; ++r) {
    int m = half * 8 + r;
    logits[(size_t)(0  + m) * VV + (v0 + mn)] = c0[r] + bias;
    logits[(size_t)(16 + m) * VV + (v0 + mn)] = c1[r] + bias;
    logits[(size_t)(32 + m) * VV + (v0 + mn)] = c2[r] + bias;
    logits[(size_t)(48 + m) * VV + (v0 + mn)] = c3[r] + bias;
  }
}

// ---------------- K6: row softmax over V (in-place) ----------------
__global__ void k_softmaxV(float* __restrict__ pv) {
  int b = blockIdx.x, tid = threadIdx.x;
  float* row = pv + (size_t)b * VV;
  __shared__ float red[256];
  float m = -1e30f;
  for (int i = tid; i < VV; i += 256) m = fmaxf(m, row[i]);
  red[tid] = m; __syncthreads();
  for (int off = 128; off > 0; off >>= 1) { if (tid < off) red[tid] = fmaxf(red[tid], red[tid + off]); __syncthreads(); }
  m = red[0]; __syncthreads();
  float s = 0.f;
  for (int i = tid; i < VV; i += 256) s += expf(row[i] - m);
  red[tid] = s; __syncthreads();
  for (int off = 128; off > 0; off >>= 1) { if (tid < off) red[tid] += red[tid + off]; __syncthreads(); }
  float inv = 1.0f / red[0];
  for (int i = tid; i < VV; i += 256) row[i] = expf(row[i] - m) * inv;
}

// ---------------- K7a: p_final = p_gen * p_vocab (+ zero tail) ----------------
__global__ void k_pfinal(const float* __restrict__ pv, const float* __restrict__ pg,
                         int ext, float* __restrict__ pf) {
  long idx = (long)blockIdx.x * blockDim.x + threadIdx.x;
  if (idx >= (long)BB * ext) return;
  int b = (int)(idx / ext);
  int j = (int)(idx - (long)b * ext);
  pf[idx] = (j < VV) ? pg[b] * pv[(size_t)b * VV + j] : 0.0f;
}

// ---------------- K7b: scatter-add copy distribution ----------------
__global__ void k_scatter(const float* __restrict__ att, const float* __restrict__ pg,
                          const int* __restrict__ fiv, int ext, float* __restrict__ pf) {
  int idx = blockIdx.x * blockDim.x + threadIdx.x;
  if (idx >= BB * SS) return;
  int b = idx / SS;
  atomicAdd(&pf[(size_t)b * ext + fiv[idx]], (1.0f - pg[b]) * att[idx]);
}

// ---------------- K8: concatenations ----------------
__global__ void k_cath(const float* __restrict__ pdh, const float* __restrict__ h,
                       float* __restrict__ outp) {
  int idx = blockIdx.x * blockDim.x + threadIdx.x;
  if (idx >= BB * (TT + 1) * HH) return;
  int b = idx / ((TT + 1) * HH);
  int rem = idx - b * (TT + 1) * HH;
  int t = rem / HH, j = rem - t * HH;
  outp[idx] = (t < TT) ? pdh[((size_t)b * TT + t) * HH + j] : h[(size_t)b * HH + j];
}

__global__ void k_catatt(const float* __restrict__ pratt, const float* __restrict__ ad,
                         float* __restrict__ outp) {
  int idx = blockIdx.x * blockDim.x + threadIdx.x;
  if (idx >= BB * (TT + 1) * SS) return;
  int b = idx / ((TT + 1) * SS);
  int rem = idx - b * (TT + 1) * SS;
  int t = rem / SS, s = rem - t * SS;
  outp[idx] = (t < TT) ? pratt[((size_t)b * TT + t) * SS + s] : ad[(size_t)b * SS + s];
}

extern "C" void kernel_launch(void* const* d_in, const int* in_sizes, int n_in,
                              void* d_out, int out_size, void* d_ws, size_t ws_size,
                              hipStream_t stream) {
  (void)in_sizes; (void)n_in; (void)ws_size;
  const int*   tok   = (const int*)d_in[0];
  const float* pdh   = (const float*)d_in[1];
  const float* lasth = (const float*)d_in[2];
  const float* enc   = (const float*)d_in[3];
  const int*   fiv   = (const int*)d_in[4];
  const float* pratt = (const float*)d_in[5];
  // d_in[6] = nb_unk_tokens scalar; ext derived from out_size instead
  const float* emb   = (const float*)d_in[7];
  const float* W_ih  = (const float*)d_in[8];
  const float* W_hh  = (const float*)d_in[9];
  const float* b_ih  = (const float*)d_in[10];
  const float* b_hh  = (const float*)d_in[11];
  const float* w_h   = (const float*)d_in[12];
  const float* w_d   = (const float*)d_in[13];
  const float* W_oh  = (const float*)d_in[14];
  const float* b_oh  = (const float*)d_in[15];
  const float* W_ov  = (const float*)d_in[16];
  const float* b_ov  = (const float*)d_in[17];
  const float* W_gen = (const float*)d_in[18];
  const float* b_gen = (const float*)d_in[19];

  long fixed = (long)BB + (long)BB * VV + (long)BB * SS +
               (long)BB * (TT + 1) * HH + (long)BB * HH + (long)BB * (TT + 1) * SS;
  int ext = (int)(((long)out_size - fixed) / BB);   // = V + nb_unk + 200

  float* out = (float*)d_out;
  float* p_final = out;  out += (size_t)BB * ext;
  float* p_gen   = out;  out += BB;
  float* p_vocab = out;  out += (size_t)BB * VV;    // staging for logits, then softmax in place
  float* att     = out;  out += (size_t)BB * SS;
  float* dec_h   = out;  out += (size_t)BB * (TT + 1) * HH;
  float* h_out   = out;  out += (size_t)BB * HH;
  float* new_pa  = out;

  float* ws = (float*)d_ws;
  float* att_dist = ws;  ws += (size_t)BB * SS;
  float* temporal = ws;  ws += (size_t)BB * SS;
  float* combined = ws;  ws += (size_t)BB * H3;
  float* ohe      = ws;

  k_gru<<<(BB * HH + 255) / 256, 256, 0, stream>>>(tok, emb, lasth, W_ih, W_hh, b_ih, b_hh, h_out);
  k_attdist<<<(BB * SS * 32 + 255) / 256, 256, 0, stream>>>(h_out, w_h, enc, pratt, att_dist, temporal);
  k_attnorm<<<BB, 256, 0, stream>>>(temporal, att);
  k_ctx<<<BB, 256, 0, stream>>>(h_out, att, enc, pdh, w_d, W_gen, b_gen, combined, p_gen);
  k_oh<<<(BB * EE + 255) / 256, 256, 0, stream>>>(combined, W_oh, b_oh, ohe);
  k_logits<<<VV / 16, 32, 0, stream>>>(ohe, W_ov, b_ov, p_vocab);
  k_softmaxV<<<BB, 256, 0, stream>>>(p_vocab);
  k_pfinal<<<(int)(((long)BB * ext + 255) / 256), 256, 0, stream>>>(p_vocab, p_gen, ext, p_final);
  k_scatter<<<(BB * SS + 255) / 256, 256, 0, stream>>>(att, p_gen, fiv, ext, p_final);
  k_cath<<<(BB * (TT + 1) * HH + 255) / 256, 256, 0, stream>>>(pdh, h_out, dec_h);
  k_catatt<<<(BB * (TT + 1) * SS + 255) / 256, 256, 0, stream>>>(pratt, att_dist, new_pa);
}